// SimGRew_31971736551537
// MI455X (gfx1250) — compile-verified
//
#include <hip/hip_runtime.h>

// -------- CDNA5 WMMA fragment types (wave32) --------
typedef __attribute__((ext_vector_type(2))) float v2f;   // 16x4 / 4x16 f32 fragment: 64 elems / 32 lanes
typedef __attribute__((ext_vector_type(8))) float v8f;   // 16x16 f32 C/D: 256 elems / 32 lanes

namespace {
constexpr int N_NODES = 4096;
constexpr int F_DIM   = 128;
constexpr int H_DIM   = 128;
constexpr int NPG     = 512;
constexpr int N_EDGES = 131072;
constexpr float ALPHA = 0.5f;

// workspace layout (float-indexed)
constexpr size_t WS_XHAT = 0;                                // N*H floats   (2 MB)
constexpr size_t WS_A0   = (size_t)N_NODES * H_DIM;          // NPG*NPG floats (1 MB) — graph-0 adjacency counts
constexpr size_t WS_NORM = WS_A0 + (size_t)NPG * NPG;        // 1 float  : sum(X_hat^2)
constexpr size_t WS_NNZ  = WS_NORM + 1;                      // 1 uint   : nonzero count of Wmat
} // namespace

// ---------------------------------------------------------------------------
// Kernel 1: X_hat = x @ W0^T + b0   (4096x128x128, fp32 WMMA 16x16x4)
// One wave per 16x16 output tile; 2048 tiles = 512 blocks * 4 waves.
// Also accumulates sum of squares into normAcc.
// ---------------------------------------------------------------------------
__global__ __launch_bounds__(128)
void xhat_wmma_kernel(const float* __restrict__ x, const float* __restrict__ W0,
                      const float* __restrict__ b0, float* __restrict__ Xhat,
                      float* __restrict__ normAcc) {
  const int lane  = threadIdx.x & 31;
  const int tile  = blockIdx.x * 4 + (threadIdx.x >> 5);   // [0, 2048)
  const int rbase = (tile >> 3) * 16;                      // node-row tile
  const int cbase = (tile & 7) * 16;                       // hidden-col tile
  const int mn    = lane & 15;                             // A: row M ; B: col N
  const int half  = lane >> 4;
  const int kb    = half * 2;                              // K sub-pair per half-wave

  v8f acc = {};
  #pragma unroll 4
  for (int k0 = 0; k0 < F_DIM; k0 += 4) {
    v2f a, b;
    const float* ap = x  + (size_t)(rbase + mn) * F_DIM + k0 + kb;  // A[m, k]
    a.x = ap[0]; a.y = ap[1];
    const float* bp = W0 + (size_t)(cbase + mn) * F_DIM + k0 + kb;  // B[k, n] = W0[n_glob, k]
    b.x = bp[0]; b.y = bp[1];
    acc = __builtin_amdgcn_wmma_f32_16x16x4_f32(false, a, false, b,
                                                (short)0, acc, false, false);
  }

  // C/D layout: VGPR v, lane l -> M = v + 8*(l>>4), N = l&15  (all 8 share N)
  const int   h    = cbase + mn;
  const float bias = b0[h];
  float ss = 0.f;
  #pragma unroll
  for (int v = 0; v < 8; ++v) {
    const float val = acc[v] + bias;
    Xhat[(size_t)(rbase + v + 8 * half) * H_DIM + h] = val;
    ss += val * val;
  }
  // wave32 reduction, then one atomic per wave
  for (int off = 16; off > 0; off >>= 1) ss += __shfl_xor(ss, off, 32);
  if (lane == 0) atomicAdd(normAcc, ss);
}

// ---------------------------------------------------------------------------
// Kernel 2: scatter edge counts. Only graph-0 edges (src < 512) can ever
// reach the output due to the local-indexing quirk of to_dense_adj.
// ---------------------------------------------------------------------------
__global__ __launch_bounds__(256)
void adj_kernel(const int* __restrict__ edge, float* __restrict__ A0) {
  const int e = blockIdx.x * blockDim.x + threadIdx.x;
  if (e >= N_EDGES) return;
  const int s = edge[e];            // src row of edge_index
  const int d = edge[N_EDGES + e];  // dst row
  if (s < NPG && d < NPG) atomicAdd(&A0[(size_t)s * NPG + d], 1.0f);
}

// ---------------------------------------------------------------------------
// Kernel 3: diagonal blocks of Wmat.
// For graph g: S = Xg @ Xg^T / norm ; W = relu(S - prob + alpha*A0[g==0]).
// One wave per 16x16 tile; 8 graphs * 32*32 tiles = 8192 waves = 1024 blocks.
// ---------------------------------------------------------------------------
__global__ __launch_bounds__(256)
void diag_wmma_kernel(const float* __restrict__ Xhat, const float* __restrict__ A0,
                      const float* __restrict__ normAcc, const float* __restrict__ prob,
                      float* __restrict__ out, unsigned int* __restrict__ nnz) {
  const int lane = threadIdx.x & 31;
  const int wave = blockIdx.x * 8 + (threadIdx.x >> 5);  // [0, 8192)
  const int g    = wave >> 10;                           // graph id
  const int t    = wave & 1023;                          // tile within 512x512 block
  const int tr   = t >> 5, tc = t & 31;
  const int gbase = g * NPG;
  const int rbase = gbase + tr * 16;
  const int cbase = gbase + tc * 16;
  const int mn   = lane & 15;
  const int half = lane >> 4;
  const int kb   = half * 2;

  v8f acc = {};
  #pragma unroll 4
  for (int k0 = 0; k0 < H_DIM; k0 += 4) {
    v2f a, b;
    const float* ap = Xhat + (size_t)(rbase + mn) * H_DIM + k0 + kb;  // A[m,k] = Xhat[row, k]
    a.x = ap[0]; a.y = ap[1];
    const float* bp = Xhat + (size_t)(cbase + mn) * H_DIM + k0 + kb;  // B[k,n] = Xhat[col, k]
    b.x = bp[0]; b.y = bp[1];
    acc = __builtin_amdgcn_wmma_f32_16x16x4_f32(false, a, false, b,
                                                (short)0, acc, false, false);
  }

  const float invNorm = 1.0f / normAcc[0];
  const float p  = prob[0];
  const int   lc = tc * 16 + mn;              // local col within graph block
  unsigned int cnt = 0;
  #pragma unroll
  for (int v = 0; v < 8; ++v) {
    const int lr = tr * 16 + v + 8 * half;    // local row within graph block
    float w = acc[v] * invNorm - p;
    if (g == 0) w += ALPHA * A0[(size_t)lr * NPG + lc];   // adjacency hits block 0 only
    w = w > 0.f ? w : 0.f;
    out[(size_t)(gbase + lr) * N_NODES + (gbase + lc)] = w;
    cnt += (w > 0.f) ? 1u : 0u;
  }
  for (int off = 16; off > 0; off >>= 1) cnt += __shfl_xor(cnt, off, 32);
  if (lane == 0) atomicAdd(nnz, cnt);
}

// ---------------------------------------------------------------------------
// Kernel 4: scalar tail: edge_ratio and prob pass-through.
// ---------------------------------------------------------------------------
__global__ void finalize_kernel(const unsigned int* __restrict__ nnz,
                                const float* __restrict__ prob,
                                float* __restrict__ out) {
  out[(size_t)N_NODES * N_NODES]     = (float)(*nnz) / (float)N_EDGES;
  out[(size_t)N_NODES * N_NODES + 1] = prob[0];
}

// ---------------------------------------------------------------------------
extern "C" void kernel_launch(void* const* d_in, const int* in_sizes, int n_in,
                              void* d_out, int out_size, void* d_ws, size_t ws_size,
                              hipStream_t stream) {
  const float* x    = (const float*)d_in[0];
  const float* W0   = (const float*)d_in[1];
  const float* b0   = (const float*)d_in[2];
  const float* prob = (const float*)d_in[3];
  const int*   edge = (const int*)d_in[4];
  // d_in[5] = batch: structure is static (node i -> graph i/512), unused.

  float* ws      = (float*)d_ws;
  float* Xhat    = ws + WS_XHAT;
  float* A0      = ws + WS_A0;
  float* normAcc = ws + WS_NORM;
  unsigned int* nnz = (unsigned int*)(ws + WS_NNZ);
  float* out     = (float*)d_out;

  // Zero adjacency + counters, and the whole output (off-diagonal stays 0).
  hipMemsetAsync((void*)A0, 0, ((size_t)NPG * NPG + 2) * sizeof(float), stream);
  hipMemsetAsync(d_out, 0, (size_t)out_size * sizeof(float), stream);

  xhat_wmma_kernel<<<512, 128, 0, stream>>>(x, W0, b0, Xhat, normAcc);
  adj_kernel<<<N_EDGES / 256, 256, 0, stream>>>(edge, A0);
  diag_wmma_kernel<<<1024, 256, 0, stream>>>(Xhat, A0, normAcc, prob, out, nnz);
  finalize_kernel<<<1, 1, 0, stream>>>(nnz, prob, out);
}